// RoIHeadTemplate_63831803953812
// MI455X (gfx1250) — compile-verified
//
#include <hip/hip_runtime.h>
#include <stdint.h>

// Problem constants (match reference)
#define B_      4
#define N_      32768
#define PRE_    4096
#define POST_   512
#define THRESH_ 0.7f
#define MASKW_  128     // PRE_/32 suppression words per row

typedef __attribute__((ext_vector_type(2))) float v2f;
typedef __attribute__((ext_vector_type(8))) float v8f;

// ---------------------------------------------------------------------------
// Kernel 1: per-anchor class max / argmax (C=3, first-occurrence tie-break)
// ---------------------------------------------------------------------------
__global__ void score_label_kernel(const float* __restrict__ cls,
                                   float* __restrict__ scores,
                                   int* __restrict__ labels) {
  int t = blockIdx.x * blockDim.x + threadIdx.x;
  if (t >= B_ * N_) return;
  const float* p = cls + (size_t)t * 3;
  float c0 = p[0], c1 = p[1], c2 = p[2];
  float best = c0; int lab = 0;
  if (c1 > best) { best = c1; lab = 1; }
  if (c2 > best) { best = c2; lab = 2; }
  scores[t] = best;
  labels[t] = lab;
}

// ---------------------------------------------------------------------------
// Kernel 2: full bitonic sort of 32768 (score,idx) keys per batch in LDS.
// 32768 * 8B = 256 KB of LDS -- fits only in CDNA5's 320 KB/WGP.
// Key = [monotone(float score) : 32][0xFFFFFFFF - idx : 32]; descending sort
// reproduces jax.lax.top_k ordering (desc score, lower index first on ties).
// Emits the top PRE_ entries + precomputed BEV box params.
// ---------------------------------------------------------------------------
__global__ void topk_sort_kernel(const float* __restrict__ scores,
                                 const float* __restrict__ boxes,
                                 int* __restrict__ topIdx,
                                 float* __restrict__ topScore,
                                 float* __restrict__ bx1, float* __restrict__ bx2,
                                 float* __restrict__ by1, float* __restrict__ by2,
                                 float* __restrict__ barea) {
  extern __shared__ unsigned long long key[];   // N_ entries (256 KB)
  const int b   = blockIdx.x;
  const int tid = threadIdx.x;
  const int nt  = blockDim.x;

  for (int t = tid; t < N_; t += nt) {
    float s = scores[b * N_ + t];
    unsigned u = __float_as_uint(s);
    u = (u & 0x80000000u) ? ~u : (u | 0x80000000u);   // order-preserving map
    key[t] = ((unsigned long long)u << 32) |
             (unsigned long long)(0xFFFFFFFFu - (unsigned)t);
  }
  __syncthreads();

  for (unsigned k = 2; k <= (unsigned)N_; k <<= 1) {
    for (unsigned j = k >> 1; j > 0; j >>= 1) {
      for (unsigned t = tid; t < (unsigned)N_; t += nt) {
        unsigned ixj = t ^ j;
        if (ixj > t) {
          unsigned long long a = key[t], c = key[ixj];
          bool descBlock = ((t & k) == 0);
          if (descBlock ? (a < c) : (a > c)) { key[t] = c; key[ixj] = a; }
        }
      }
      __syncthreads();
    }
  }

  for (int t = tid; t < PRE_; t += nt) {
    unsigned long long kk = key[t];
    unsigned u = (unsigned)(kk >> 32);
    unsigned n = 0xFFFFFFFFu - (unsigned)(kk & 0xFFFFFFFFull);
    unsigned bits = (u & 0x80000000u) ? (u & 0x7FFFFFFFu) : ~u;
    topIdx[b * PRE_ + t]   = (int)n;
    topScore[b * PRE_ + t] = __uint_as_float(bits);
    const float* bp = boxes + ((size_t)b * N_ + n) * 7;
    float x = bp[0], y = bp[1], dx = bp[3], dy = bp[4];
    bx1[b * PRE_ + t]   = x - 0.5f * dx;
    bx2[b * PRE_ + t]   = x + 0.5f * dx;
    by1[b * PRE_ + t]   = y - 0.5f * dy;
    by2[b * PRE_ + t]   = y + 0.5f * dy;
    barea[b * PRE_ + t] = dx * dy;
  }
}

// ---------------------------------------------------------------------------
// Kernel 3: suppression bitmask. One wave32 per 16-row x 32-col tile.
// area_i + area_j (union additive term) computed on the matrix unit:
//   A(16x4) = [area_i, 1, 0, 0], B(4x16) = [1; area_j; 0; 0]
//   D = V_WMMA_F32_16X16X4_F32(A,B,0) = area_i + area_j  (exact f32 FMA)
// VALU does intersection min/max + threshold; ballot_w32 packs bits.
// Each block owns 16 distinct 32-bit mask words -> plain stores.
// ---------------------------------------------------------------------------
__global__ void iou_mask_kernel(const float* __restrict__ bx1,
                                const float* __restrict__ bx2,
                                const float* __restrict__ by1,
                                const float* __restrict__ by2,
                                const float* __restrict__ barea,
                                unsigned* __restrict__ mask) {
  const int ct = blockIdx.x;            // 0..127  (32-col tiles)
  const int rt = blockIdx.y;            // 0..255  (16-row tiles)
  const int b  = blockIdx.z;
  const int lane = threadIdx.x;         // wave32
  const int rowBase = rt * 16, colBase = ct * 32;
  const int off = b * PRE_;
  const int nLoc = lane & 15;
  const bool hiHalf = (lane >= 16);

  // A fragment (16x4 f32): lanes 0-15 -> M=lane, {K0,K1}; lanes 16-31 -> {K2,K3}=0
  v2f a;
  if (!hiHalf) { a.x = barea[off + rowBase + lane]; a.y = 1.0f; }
  else         { a.x = 0.0f;                        a.y = 0.0f; }
  // B fragments (4x16 f32): lanes 0-15 -> {B[0][N], B[1][N]}; lanes 16-31 -> 0
  v2f b0, b1;
  if (!hiHalf) {
    b0.x = 1.0f; b0.y = barea[off + colBase + nLoc];
    b1.x = 1.0f; b1.y = barea[off + colBase + 16 + nLoc];
  } else {
    b0.x = 0.0f; b0.y = 0.0f; b1.x = 0.0f; b1.y = 0.0f;
  }
  v8f cz = {};
  // (neg_a, A, neg_b, B, c_mod, C, reuse_a, reuse_b)
  v8f sum0 = __builtin_amdgcn_wmma_f32_16x16x4_f32(false, a, false, b0,
                                                   (short)0, cz, false, false);
  v8f sum1 = __builtin_amdgcn_wmma_f32_16x16x4_f32(false, a, false, b1,
                                                   (short)0, cz, false, false);

  const int j0 = colBase + nLoc;
  const int j1 = j0 + 16;
  const float jx1_0 = bx1[off + j0], jx2_0 = bx2[off + j0];
  const float jy1_0 = by1[off + j0], jy2_0 = by2[off + j0];
  const float jx1_1 = bx1[off + j1], jx2_1 = bx2[off + j1];
  const float jy1_1 = by1[off + j1], jy2_1 = by2[off + j1];

  unsigned bal0[8], bal1[8];
#pragma unroll
  for (int r = 0; r < 8; ++r) {
    const int i = rowBase + r + (hiHalf ? 8 : 0);   // matches C/D VGPR layout
    const float ix1 = bx1[off + i], ix2 = bx2[off + i];
    const float iy1 = by1[off + i], iy2 = by2[off + i];

    float iw0 = fmaxf(fminf(ix2, jx2_0) - fmaxf(ix1, jx1_0), 0.0f);
    float ih0 = fmaxf(fminf(iy2, jy2_0) - fmaxf(iy1, jy1_0), 0.0f);
    float inter0 = iw0 * ih0;
    float uni0 = fmaxf(sum0[r] - inter0, 1e-6f);
    bool p0 = (inter0 > THRESH_ * uni0) && (j0 > i);

    float iw1 = fmaxf(fminf(ix2, jx2_1) - fmaxf(ix1, jx1_1), 0.0f);
    float ih1 = fmaxf(fminf(iy2, jy2_1) - fmaxf(iy1, jy1_1), 0.0f);
    float inter1 = iw1 * ih1;
    float uni1 = fmaxf(sum1[r] - inter1, 1e-6f);
    bool p1 = (inter1 > THRESH_ * uni1) && (j1 > i);

    bal0[r] = __builtin_amdgcn_ballot_w32(p0);
    bal1[r] = __builtin_amdgcn_ballot_w32(p1);
  }

  if (lane == 0) {
#pragma unroll
    for (int r = 0; r < 8; ++r) {
      unsigned lo = (bal0[r] & 0xFFFFu) | ((bal1[r] & 0xFFFFu) << 16);
      unsigned hi = (bal0[r] >> 16)     | ((bal1[r] >> 16) << 16);
      mask[(size_t)(off + rowBase + r)     * MASKW_ + ct] = lo;
      mask[(size_t)(off + rowBase + r + 8) * MASKW_ + ct] = hi;
    }
  }
}

// ---------------------------------------------------------------------------
// Kernel 4: serial greedy scan. One 128-thread block per batch.
// ---------------------------------------------------------------------------
__global__ void nms_scan_kernel(const unsigned* __restrict__ mask,
                                int* __restrict__ sel,
                                int* __restrict__ cnt) {
  const int b = blockIdx.x;
  const int tid = threadIdx.x;      // 0..127, owns word tid
  __shared__ unsigned remv[MASKW_];
  __shared__ int flag;
  __shared__ int scnt;
  remv[tid] = 0u;
  if (tid == 0) scnt = 0;
  __syncthreads();

  for (int i = 0; i < PRE_; ++i) {
    const int w = i >> 5;
    const unsigned bitm = 1u << (i & 31);
    if (tid == 0) flag = ((remv[w] & bitm) == 0u) ? 1 : 0;
    __syncthreads();
    if (flag) {
      remv[tid] |= mask[(size_t)(b * PRE_ + i) * MASKW_ + tid];
      if (tid == 0 && scnt < POST_) { sel[b * POST_ + scnt] = i; scnt++; }
    }
    __syncthreads();
  }
  if (tid == 0) cnt[b] = scnt;
}

// ---------------------------------------------------------------------------
// Kernel 5: gather outputs. out = [rois(4*512*7) | scores(4*512) | labels(4*512)]
// Invalid slots: rois=0, score=0, label=1 (reference: where(valid,...)+1).
// ---------------------------------------------------------------------------
__global__ void output_kernel(const float* __restrict__ boxes,
                              const float* __restrict__ scores,
                              const int* __restrict__ labels,
                              const int* __restrict__ topIdx,
                              const int* __restrict__ sel,
                              const int* __restrict__ cnt,
                              float* __restrict__ out) {
  const int b = blockIdx.x;
  const int s = threadIdx.x;        // 0..511
  float roi[7] = {0, 0, 0, 0, 0, 0, 0};
  float sc = 0.0f;
  float lb = 1.0f;
  if (s < cnt[b]) {
    const int i = sel[b * POST_ + s];
    const int orig = topIdx[b * PRE_ + i];
    const float* bp = boxes + ((size_t)b * N_ + orig) * 7;
#pragma unroll
    for (int k = 0; k < 7; ++k) roi[k] = bp[k];
    sc = scores[b * N_ + orig];
    lb = (float)(labels[b * N_ + orig] + 1);
  }
  float* roisOut   = out;
  float* scoresOut = out + (size_t)B_ * POST_ * 7;
  float* labelsOut = out + (size_t)B_ * POST_ * 7 + (size_t)B_ * POST_;
#pragma unroll
  for (int k = 0; k < 7; ++k) roisOut[((size_t)b * POST_ + s) * 7 + k] = roi[k];
  scoresOut[b * POST_ + s] = sc;
  labelsOut[b * POST_ + s] = lb;
}

// ---------------------------------------------------------------------------
extern "C" void kernel_launch(void* const* d_in, const int* in_sizes, int n_in,
                              void* d_out, int out_size, void* d_ws, size_t ws_size,
                              hipStream_t stream) {
  const float* boxes = (const float*)d_in[0];   // (4,32768,7)
  const float* cls   = (const float*)d_in[1];   // (4,32768,3)
  float* out = (float*)d_out;

  // Workspace carving (all 4-byte aligned)
  char* ws = (char*)d_ws;
  float*    scores   = (float*)ws;                 ws += sizeof(float) * B_ * N_;
  int*      labels   = (int*)ws;                   ws += sizeof(int)   * B_ * N_;
  int*      topIdx   = (int*)ws;                   ws += sizeof(int)   * B_ * PRE_;
  float*    topScore = (float*)ws;                 ws += sizeof(float) * B_ * PRE_;
  float*    bx1      = (float*)ws;                 ws += sizeof(float) * B_ * PRE_;
  float*    bx2      = (float*)ws;                 ws += sizeof(float) * B_ * PRE_;
  float*    by1      = (float*)ws;                 ws += sizeof(float) * B_ * PRE_;
  float*    by2      = (float*)ws;                 ws += sizeof(float) * B_ * PRE_;
  float*    barea    = (float*)ws;                 ws += sizeof(float) * B_ * PRE_;
  unsigned* mask     = (unsigned*)ws;              ws += sizeof(unsigned) * (size_t)B_ * PRE_ * MASKW_;
  int*      sel      = (int*)ws;                   ws += sizeof(int) * B_ * POST_;
  int*      cnt      = (int*)ws;

  // 1) class max / argmax
  score_label_kernel<<<(B_ * N_ + 255) / 256, 256, 0, stream>>>(cls, scores, labels);

  // 2) per-batch in-LDS bitonic sort (256 KB dynamic LDS per workgroup)
  topk_sort_kernel<<<B_, 1024, (size_t)N_ * sizeof(unsigned long long), stream>>>(
      scores, boxes, topIdx, topScore, bx1, bx2, by1, by2, barea);

  // 3) IoU suppression bitmask (WMMA for union additive term)
  dim3 gridIou(PRE_ / 32, PRE_ / 16, B_);
  iou_mask_kernel<<<gridIou, 32, 0, stream>>>(bx1, bx2, by1, by2, barea, mask);

  // 4) serial greedy scan, one block per batch
  nms_scan_kernel<<<B_, MASKW_, 0, stream>>>(mask, sel, cnt);

  // 5) gather outputs
  output_kernel<<<B_, POST_, 0, stream>>>(boxes, scores, labels, topIdx, sel, cnt, out);
}